// RamaFuse_9225589752448
// MI455X (gfx1250) — compile-verified
//
#include <hip/hip_runtime.h>
#include <math.h>

// Problem constants (fixed by the reference)
#define BB 8
#define TT 512
#define MM 8
#define DD 128
#define QQ 16
#define WW 16

typedef __attribute__((ext_vector_type(2))) float v2f;
typedef __attribute__((ext_vector_type(8))) float v8f;
typedef __attribute__((ext_vector_type(4))) unsigned int v4u;
typedef __attribute__((ext_vector_type(8))) int v8i;
typedef __attribute__((ext_vector_type(4))) int v4i;

// ---------------------------------------------------------------------------
// Ramanujan FIR coefficients (row for period q, 16 taps), computed in fp64 to
// match the numpy float64 reference path, then cast to fp32.
// ---------------------------------------------------------------------------
__device__ __forceinline__ void rama_row(int q, float* out16) {
    double c[16];
    for (int w = 0; w < 16; ++w) c[w] = 0.0;
    int cnt = 0;
    for (int a = 1; a <= q; ++a) {
        int x = a, y = q;
        while (y) { int t = x % y; x = y; y = t; }
        if (x == 1) {
            ++cnt;
            double f = 2.0 * 3.14159265358979323846 * (double)a / (double)q;
            for (int w = 0; w < 16; ++w) c[w] += cos(f * (double)w);
        }
    }
    if (cnt == 0) { for (int w = 0; w < 16; ++w) out16[w] = 1.0f; return; }
    double mean = 0.0;
    for (int w = 0; w < 16; ++w) mean += c[w];
    mean *= (1.0 / 16.0);
    double ss = 0.0;
    for (int w = 0; w < 16; ++w) { c[w] -= mean; ss += c[w] * c[w]; }
    double denom = sqrt(ss > 1e-6 ? ss : 1e-6);
    for (int w = 0; w < 16; ++w) out16[w] = (float)(c[w] / denom);
}

// ---------------------------------------------------------------------------
// TDM: load a 2D f32 tile [rows=16, cols=256] (row stride 512 elements) from
// global memory into LDS at byte offset lds_off. D# per CDNA5 ISA ch.8:
//   group0: count=1, lds_addr, global_addr(57b), type=2
//   group1: data_size=4B, tensor 512x16, tile 256x16, dim0 stride 512
//   groups 2/3: zero (2-D tensor -> disabled)
// Toolchain uses the 6-arg builtin form (g0, g1, g2, g3, extra v8i, cpol).
// ---------------------------------------------------------------------------
__device__ __forceinline__ void tdm_load_g_tile(const float* gsrc_tile,
                                                unsigned lds_off) {
    unsigned long long ga = (unsigned long long)(size_t)gsrc_tile;
    v4u g0;
    g0[0] = 1u;                                            // count=1, user D#
    g0[1] = lds_off;                                       // lds_addr (bytes)
    g0[2] = (unsigned)(ga & 0xFFFFFFFFu);                  // global_addr lo
    g0[3] = (unsigned)((ga >> 32) & 0x01FFFFFFu) | (2u << 30); // addr hi|type=2
    v8i g1;
    g1[0] = (int)(2u << 16);        // wg_mask=0 | data_size=2 (4 bytes)
    g1[1] = (int)(512u << 16);      // atomic_barrier=0 | tensor_dim0[15:0]=512
    g1[2] = (int)(16u << 16);       // tensor_dim0 hi=0 | tensor_dim1[15:0]=16
    g1[3] = (int)(256u << 16);      // tensor_dim1 hi=0 | tile_dim0=256
    g1[4] = 16;                     // tile_dim1=16 | tile_dim2=0
    g1[5] = 512;                    // tensor_dim0_stride lo = 512
    g1[6] = 0;                      // stride hi | tensor_dim1_stride lo
    g1[7] = 0;                      // tensor_dim1_stride hi
    v4i z4 = {0, 0, 0, 0};
    v8i z8 = {0, 0, 0, 0, 0, 0, 0, 0};
    __builtin_amdgcn_tensor_load_to_lds(g0, g1, z4, z4, z8, 0);
}

// ---------------------------------------------------------------------------
// Kernel 1: z_anl[b,t,m] = mean_d(z[b,t,m,:]) * vm[b,t,m]
// One wave per row; coalesced float4 reads + butterfly reduce.
// rows = B*T*M = 32768 -> 4096 blocks of 256 threads.
// ---------------------------------------------------------------------------
__global__ void __launch_bounds__(256) k_anl(const float* __restrict__ z,
                                             const float* __restrict__ vm,
                                             float* __restrict__ za) {
    int gw   = (blockIdx.x * 256 + threadIdx.x) >> 5;   // row index
    int lane = threadIdx.x & 31;
    const float4* zr = (const float4*)z + (size_t)gw * 32;  // 128 floats
    float4 v = zr[lane];
    float s = v.x + v.y + v.z + v.w;
#pragma unroll
    for (int off = 16; off; off >>= 1) s += __shfl_xor(s, off, 32);
    if (lane == 0) za[gw] = s * (1.0f / 128.0f) * vm[gw];
}

// ---------------------------------------------------------------------------
// Kernel 2: analysis FIR + gate MLP with the torch-view index scramble.
// One WG per (b, mg): needs r[b, q'=2mg..2mg+1, m'=0..7, :].
// g_out[((b*8+mg)*16 + p)*512 + t]
// ---------------------------------------------------------------------------
__global__ void __launch_bounds__(256) k_gate(const float* __restrict__ za_g,
                                              const float* __restrict__ w1,
                                              const float* __restrict__ b1,
                                              const float* __restrict__ w2,
                                              const float* __restrict__ b2,
                                              float* __restrict__ g_out) {
    __shared__ float za[8 * 528];    // per-m' padded series (15 left pad)
    __shared__ float klds[256];      // K[q][w]
    __shared__ float w1s[16], b1s[16], b2s[16], w2s[256];
    int tid = threadIdx.x;
    int b   = blockIdx.x >> 3;
    int mg  = blockIdx.x & 7;

    if (tid < 16) {
        rama_row(tid + 1, &klds[tid * 16]);
        w1s[tid] = w1[tid]; b1s[tid] = b1[tid]; b2s[tid] = b2[tid];
    }
    w2s[tid] = w2[tid];
    if (tid < 8 * 15) { int mp = tid / 15, i = tid % 15; za[mp * 528 + i] = 0.0f; }
#pragma unroll 4
    for (int i = 0; i < 16; ++i) {
        int flat = tid + i * 256;            // = t*8 + m'
        int t = flat >> 3, mp = flat & 7;
        za[mp * 528 + t + 15] = za_g[b * 4096 + flat];
    }
    __syncthreads();

    for (int half = 0; half < 2; ++half) {
        int t = tid + half * 256;
        float h1v[16];
#pragma unroll
        for (int qq = 0; qq < 16; ++qq) {
            int qp = 2 * mg + (qq >> 3);     // filter row (scrambled view)
            int mp = qq & 7;                 // source channel
            float r = 0.0f;
#pragma unroll
            for (int w = 0; w < 16; ++w)
                r = fmaf(klds[qp * 16 + w], za[mp * 528 + t + w], r);
            float pre = fmaf(r, w1s[qq], b1s[qq]);
            h1v[qq] = 0.5f * pre * (1.0f + erff(pre * 0.70710678118654752f));
        }
        int base = ((b * 8 + mg) * 16) * 512 + t;
#pragma unroll
        for (int p = 0; p < 16; ++p) {
            float a = b2s[p];
#pragma unroll
            for (int qq = 0; qq < 16; ++qq) a = fmaf(w2s[p * 16 + qq], h1v[qq], a);
            g_out[base + p * 512] = 1.0f / (1.0f + __expf(-a));
        }
    }
}

// ---------------------------------------------------------------------------
// Kernel 3: synthesis FIR via V_WMMA_F32_16X16X4_F32 + gated Q-reduction.
// WG = (b, m, dblk of 32 channels, t-half of 256 steps). 512 WGs x 256 thr.
// Gate tile staged by the Tensor Data Mover (tensor_load_to_lds, TENSORcnt);
// masked z transposed into LDS by VALU (mask multiply fused into staging).
// D tile = R[q(16), d(16)] at one t; A = K (16x16, chained over 4 k-chunks),
// B = x windows over (w, d) from LDS. Gate broadcast in C-layout, per-lane
// dot over 8 acc VGPRs, cross-half shuffle completes sum over q.
// ---------------------------------------------------------------------------
#define XSTR 273   // 256 + 15 pad + 2 spare; 273 % 64 = 17 -> conflict-free

__global__ void __launch_bounds__(256) k_syn(const float* __restrict__ z,
                                             const float* __restrict__ vm,
                                             const float* __restrict__ g_ws,
                                             const float* __restrict__ beta_p,
                                             float* __restrict__ out) {
    __shared__ float xs[32 * XSTR];  // masked z, transposed [d_local][tau]
    __shared__ float gs[16 * 256];   // g[q][t_local]  (filled by TDM)
    __shared__ float vs[256];        // vm[t_local]
    __shared__ float ks[256];        // K[q][w]

    int tid  = threadIdx.x;
    int blk  = blockIdx.x;
    int th   = blk & 1;              // t half
    int dblk = (blk >> 1) & 3;       // 32-channel block
    int m    = (blk >> 3) & 7;
    int b    = blk >> 6;
    int t0   = th * 256;
    int wv   = tid >> 5;

    // Kick off the TDM gate-tile DMA first so it overlaps the x staging.
    if (wv == 0) {
        unsigned gs_off = (unsigned)(size_t)(void*)gs;  // flat LDS: offset in [31:0]
        tdm_load_g_tile(g_ws + (size_t)(b * 8 + m) * 8192 + t0, gs_off);
    }

    if (tid < 16) rama_row(tid + 1, &ks[tid * 16]);

    // Stage masked z, transposed. Main body: 256 t x 32 d.
    {
        const float* zb  = z  + (size_t)((b * 512 + t0) * 8 + m) * 128 + dblk * 32;
        const float* vmb = vm + (size_t)(b * 512 + t0) * 8 + m;
#pragma unroll 4
        for (int i = 0; i < 32; ++i) {
            int flat = tid + i * 256;
            int tl = flat >> 5, dl = flat & 31;
            float v  = vmb[tl * 8];
            float zv = zb[(size_t)tl * 1024 + dl];
            xs[dl * XSTR + tl + 15] = zv * v;
            if (dl == 0) vs[tl] = v;
        }
        // 15-sample history head (zeros for t0==0, else previous masked z)
        if (tid < 480) {
            int dl = tid / 15, i = tid % 15;
            float val = 0.0f;
            if (t0 != 0) {
                int tg = t0 - 15 + i;
                val = z[(size_t)((b * 512 + tg) * 8 + m) * 128 + dblk * 32 + dl]
                    * vm[(size_t)(b * 512 + tg) * 8 + m];
            }
            xs[dl * XSTR + i] = val;
        }
    }
    if (wv == 0) __builtin_amdgcn_s_wait_tensorcnt(0);  // gs tile landed
    __syncthreads();

    int lane = tid & 31;
    int half = lane >> 4;       // lane half selects K-chunk parity / q-halves
    int mrow = lane & 15;       // A row (q) and B col (d) index

    // A fragments: 16x4 f32 layout -> lane's two taps w = 4c + 2*half + {0,1}
    v2f afrag[4];
#pragma unroll
    for (int c = 0; c < 4; ++c) {
        int wb = 4 * c + 2 * half;
        afrag[c].x = ks[mrow * 16 + wb];
        afrag[c].y = ks[mrow * 16 + wb + 1];
    }

    float beta = beta_p[0];
    const float* zio = z   + (size_t)((b * 512 + t0) * 8 + m) * 128 + dblk * 32;
    float*       oio = out + (size_t)((b * 512 + t0) * 8 + m) * 128 + dblk * 32;

    for (int ti = 0; ti < 32; ++ti) {
        int tl = wv * 32 + ti;           // local t within [0,256)
        // Gate operand in C-layout: VGPR v -> q = v + 8*half (broadcast reads)
        float gq[8];
#pragma unroll
        for (int v = 0; v < 8; ++v) gq[v] = gs[(v + 8 * half) * 256 + tl];

        float pd[2];
#pragma unroll
        for (int dt = 0; dt < 2; ++dt) {
            int d0 = dt * 16;
            v8f acc = {0.f, 0.f, 0.f, 0.f, 0.f, 0.f, 0.f, 0.f};
#pragma unroll
            for (int c = 0; c < 4; ++c) {
                // B 4x16 layout: VGPR j, lane half h -> row w = 4c + 2h + j
                int base = (d0 + mrow) * XSTR + tl + 4 * c + 2 * half;
                v2f bfrag;
                bfrag.x = xs[base];
                bfrag.y = xs[base + 1];
                acc = __builtin_amdgcn_wmma_f32_16x16x4_f32(
                        false, afrag[c], false, bfrag, (short)0, acc,
                        false, false);
            }
            // Per-lane partial: sum over this half's 8 q-rows, gated
            float s = acc[0] * gq[0] + acc[1] * gq[1] + acc[2] * gq[2] +
                      acc[3] * gq[3] + acc[4] * gq[4] + acc[5] * gq[5] +
                      acc[6] * gq[6] + acc[7] * gq[7];
            s += __shfl_xor(s, 16, 32);   // combine q=0..7 with q=8..15
            pd[dt] = s;
        }
        // lane l writes d_local = l (tile 0 for l<16, tile 1 for l>=16)
        float pv = half ? pd[1] : pd[0];
        float zv = zio[(size_t)tl * 1024 + lane];
        oio[(size_t)tl * 1024 + lane] = fmaf(beta * vs[tl], pv, zv);
    }
}

// ---------------------------------------------------------------------------
extern "C" void kernel_launch(void* const* d_in, const int* in_sizes, int n_in,
                              void* d_out, int out_size, void* d_ws, size_t ws_size,
                              hipStream_t stream) {
    (void)in_sizes; (void)n_in; (void)out_size; (void)ws_size;
    const float* z    = (const float*)d_in[0];
    const float* vmk  = (const float*)d_in[1];
    const float* w1   = (const float*)d_in[2];
    const float* b1   = (const float*)d_in[3];
    const float* w2   = (const float*)d_in[4];
    const float* b2   = (const float*)d_in[5];
    const float* beta = (const float*)d_in[6];
    float* out = (float*)d_out;

    float* za = (float*)d_ws;          // 32768 floats  (z_anl [B,T,M])
    float* g  = za + 32768;            // 524288 floats (g [B*M,Q,T])

    k_anl <<<4096, 256, 0, stream>>>(z, vmk, za);
    k_gate<<<64,   256, 0, stream>>>(za, w1, b1, w2, b2, g);
    k_syn <<<512,  256, 0, stream>>>(z, vmk, g, beta, out);
}